// HGCNmodel2_77163382440889
// MI455X (gfx1250) — compile-verified
//
#include <hip/hip_runtime.h>

// ---------------------------------------------------------------------------
// HGCN 2-hop sparse aggregation for MI455X (gfx1250, wave32).
//
//   h_i  = relu(spmm(A_i, W))     (A_i: COO, E=1.6M edges, N=100k, D=64)
//   out_i = spmm(A_i, h_i)
//
// Memory-bound (~200 MB effective traffic @ 23.3 TB/s). Strategy:
//   * scatter-add SpMM with f32 global atomics (L2-resident outputs)
//   * relu folded into the gather of the second hop (no relu pass)
//   * the exact per-edge scaling  Y[rows[e]] += vals[e]*X[cols[e]]  is
//     expressed per 16-edge group as  C = sum_s A_s(16x4) x B_s(4x16)
//     with A_s = K-slice of diag(vals)  ->  V_WMMA_F32_16X16X4_F32 (exact
//     f32 WMMA), 4 col-tiles of 16 to cover D=64.
//
// Layouts per CDNA5 ISA 7.12.2 (wave32):
//   A 16x4 f32 : lane l holds M=l%16; VGPR0 = K=2*(l>=16), VGPR1 = K+1
//   B 4x16 f32 : lane l holds N=l%16; VGPR0 = K=2*(l>=16), VGPR1 = K+1
//   C 16x16 f32: VGPR j, lane l -> (M = j + 8*(l>=16), N = l%16)
// ---------------------------------------------------------------------------

typedef __attribute__((ext_vector_type(2))) float v2f;
typedef __attribute__((ext_vector_type(8))) float v8f;

#define DOUT 64

__global__ void zero_f4(float4* __restrict__ p, long n4) {
  long i = (long)blockIdx.x * blockDim.x + threadIdx.x;
  long stride = (long)gridDim.x * blockDim.x;
  for (; i < n4; i += stride) p[i] = make_float4(0.f, 0.f, 0.f, 0.f);
}

template <bool RELU>
__global__ __launch_bounds__(256) void spmm_wmma(
    const int* __restrict__ rows, const int* __restrict__ cols,
    const float* __restrict__ vals, const float* __restrict__ X,
    float* __restrict__ Y, int ngroups)
{
  const int gid  = blockIdx.x * blockDim.x + threadIdx.x;
  const int wave = gid >> 5;
  if (wave >= ngroups) return;          // wave-uniform: EXEC stays all-1s
  const int lane = threadIdx.x & 31;
  const int half = lane >> 4;           // 0 or 1
  const int n    = lane & 15;           // tile column, and "my" M slot
  const long base = (long)wave << 4;    // 16 edges per group

  // A fragments: diag(vals[base..base+15]) cut into 4 K-slices of 16x4.
  // A_s[m,k] = vals[m] if m == 4s+k else 0.
  const float vm = vals[base + n];
  v2f a[4];
#pragma unroll
  for (int s = 0; s < 4; ++s) {
    a[s].x = (n == 4 * s + 2 * half)     ? vm : 0.f;
    a[s].y = (n == 4 * s + 2 * half + 1) ? vm : 0.f;
  }

  // Gather column indices per K-slice (uniform per half-wave -> broadcast ld).
  int c0[4], c1[4];
#pragma unroll
  for (int s = 0; s < 4; ++s) {
    c0[s] = cols[base + 4 * s + 2 * half];
    c1[s] = cols[base + 4 * s + 2 * half + 1];
  }

  // Destination rows for my 8 C-slots (M = j + 8*half).
  int r[8];
#pragma unroll
  for (int j = 0; j < 8; ++j) r[j] = rows[base + j + 8 * half];

#pragma unroll
  for (int t = 0; t < 4; ++t) {         // 4 column tiles of 16 -> D=64
    v8f acc = {};
#pragma unroll
    for (int s = 0; s < 4; ++s) {
      // B_s[k, n] = X[cols[base+4s+k]][16t + n]  (relu'd on 2nd hop).
      float bx = X[(long)c0[s] * DOUT + 16 * t + n];
      float by = X[(long)c1[s] * DOUT + 16 * t + n];
      if (RELU) { bx = fmaxf(bx, 0.f); by = fmaxf(by, 0.f); }
      v2f b; b.x = bx; b.y = by;
      // Exact f32 MMA: C[m,:] += vals[m] * X[cols[m], tile]
      acc = __builtin_amdgcn_wmma_f32_16x16x4_f32(
          /*neg_a=*/false, a[s], /*neg_b=*/false, b,
          /*c_mod=*/(short)0, acc, /*reuse_a=*/false, /*reuse_b=*/false);
    }
    // Scatter C tile: segment-sum via f32 global atomics (L2).
#pragma unroll
    for (int j = 0; j < 8; ++j) {
      atomicAdd(&Y[(long)r[j] * DOUT + 16 * t + n], acc[j]);
    }
  }
}

// Scalar fallback for a non-multiple-of-16 edge tail (E=1.6M -> unused).
__global__ void spmm_tail(const int* __restrict__ rows, const int* __restrict__ cols,
                          const float* __restrict__ vals, const float* __restrict__ X,
                          float* __restrict__ Y, int start, int count, int relu)
{
  int tid = blockIdx.x * blockDim.x + threadIdx.x;
  if (tid >= count * 16) return;
  long e = start + (tid >> 4);
  int  f = (tid & 15) * 4;
  int  rr = rows[e], cc = cols[e];
  float v = vals[e];
  const float* xp = X + (long)cc * DOUT + f;
  float*       yp = Y + (long)rr * DOUT + f;
#pragma unroll
  for (int i = 0; i < 4; ++i) {
    float x = xp[i];
    if (relu) x = fmaxf(x, 0.f);
    atomicAdd(&yp[i], v * x);
  }
}

static inline void run_spmm(const int* r, const int* c, const float* v, int E,
                            const float* X, float* Y, bool relu, hipStream_t stream)
{
  int ngroups = E / 16;
  int rem = E - ngroups * 16;
  if (ngroups > 0) {
    int blocks = (ngroups * 32 + 255) / 256;
    if (relu) spmm_wmma<true ><<<blocks, 256, 0, stream>>>(r, c, v, X, Y, ngroups);
    else      spmm_wmma<false><<<blocks, 256, 0, stream>>>(r, c, v, X, Y, ngroups);
  }
  if (rem > 0) {
    int blocks = (rem * 16 + 255) / 256;
    spmm_tail<<<blocks, 256, 0, stream>>>(r, c, v, X, Y, ngroups * 16, rem, relu ? 1 : 0);
  }
}

extern "C" void kernel_launch(void* const* d_in, const int* in_sizes, int n_in,
                              void* d_out, int out_size, void* d_ws, size_t ws_size,
                              hipStream_t stream)
{
  const int*   a1_rows = (const int*)  d_in[0];
  const int*   a1_cols = (const int*)  d_in[1];
  const float* a1_vals = (const float*)d_in[2];
  const int*   a2_rows = (const int*)  d_in[3];
  const int*   a2_cols = (const int*)  d_in[4];
  const float* a2_vals = (const float*)d_in[5];
  const float* W       = (const float*)d_in[6];

  const int  E1 = in_sizes[0];
  const int  E2 = in_sizes[3];
  const long N  = (long)in_sizes[6] / DOUT;   // W is [N, 64]

  float* out1 = (float*)d_out;                // [N,64]
  float* out2 = out1 + N * DOUT;              // [N,64]
  float* h1   = (float*)d_ws;                 // [N,64] scratch
  float* h2   = h1 + N * DOUT;                // [N,64] scratch

  // Zero accumulators (h1,h2 in ws; out1,out2 in d_out).
  zero_f4<<<4096, 256, 0, stream>>>((float4*)d_ws,  (2L * N * DOUT) / 4);
  zero_f4<<<4096, 256, 0, stream>>>((float4*)d_out, (long)out_size / 4);

  // Hop 1: h_i = A_i @ W   (relu deferred to hop-2 gather).
  run_spmm(a1_rows, a1_cols, a1_vals, E1, W, h1, /*relu=*/false, stream);
  run_spmm(a2_rows, a2_cols, a2_vals, E2, W, h2, /*relu=*/false, stream);

  // Hop 2: out_i = A_i @ relu(h_i)   (relu applied on the gathered B tiles).
  run_spmm(a1_rows, a1_cols, a1_vals, E1, h1, out1, /*relu=*/true, stream);
  run_spmm(a2_rows, a2_cols, a2_vals, E2, h2, out2, /*relu=*/true, stream);
}